// Rule_15599321219396
// MI455X (gfx1250) — compile-verified
//
#include <hip/hip_runtime.h>
#include <hip/hip_bf16.h>
#include <stdint.h>

// Problem geometry (fixed by the reference)
#define B_  8
#define H_  2048
#define W_  2048
#define HW_ (H_ * W_)
#define PLANE2_ (2 * HW_)                       // one batch of x/state: [s, b] planes
#define STATE_ELEMS ((size_t)B_ * PLANE2_)      // 67,108,864 floats
#define NINV (1.0f / 4194304.0f)                // 1 / (C*H*W)

typedef float v2f __attribute__((ext_vector_type(2)));
typedef float v4f __attribute__((ext_vector_type(4)));
typedef float v8f __attribute__((ext_vector_type(8)));

// ---- CDNA5 async global->LDS staging (ASYNCcnt path) -----------------------
// the LDS address is taken from the vdst VGPR (plus offset); generic shared pointers carry the
// wave-relative LDS offset in their low 32 bits, so truncation is the offset.
__device__ __forceinline__ void async_ld_b128(uint32_t lds_off, const void* gaddr) {
    asm volatile("global_load_async_to_lds_b128 %0, %1, off"
                 :: "v"(lds_off), "v"(gaddr) : "memory");
}
__device__ __forceinline__ void wait_async0() {
    asm volatile("s_wait_asynccnt 0" ::: "memory");
}

// ---- main fused kernel: one block per (batch, row) -------------------------
__global__ __launch_bounds__(256) void ising_row_kernel(
    const float* __restrict__ x, const float* __restrict__ rnd,
    const float* __restrict__ dropr, float* __restrict__ out,
    float* __restrict__ partials)
{
    __shared__ __align__(16) float ls[3][W_];   // up / center / down rows (24 KB)
    __shared__ float rede[256];
    __shared__ float redm[256];

    const int tid = threadIdx.x;
    const int blk = blockIdx.x;
    const int bi  = blk >> 11;                  // blk / 2048
    const int h   = blk & (H_ - 1);
    const int hu  = (h + H_ - 1) & (H_ - 1);    // roll(+1) neighbor: s[h-1]
    const int hd  = (h + 1) & (H_ - 1);         // roll(-1) neighbor: s[h+1]

    const float* sp  = x + (size_t)bi * PLANE2_;
    const float* s_u = sp + (size_t)hu * W_;
    const float* s_c = sp + (size_t)h  * W_;
    const float* s_d = sp + (size_t)hd * W_;

    // Stage 3 rows into LDS via async b128 loads (2 per thread per row).
    #pragma unroll
    for (int c = 0; c < 2; ++c) {
        const int w4 = (c * 256 + tid) * 4;
        async_ld_b128((uint32_t)(uintptr_t)&ls[0][w4], s_u + w4);
        async_ld_b128((uint32_t)(uintptr_t)&ls[1][w4], s_c + w4);
        async_ld_b128((uint32_t)(uintptr_t)&ls[2][w4], s_d + w4);
    }
    wait_async0();        // this wave's async writes to LDS are complete
    __syncthreads();      // make all waves' staged rows visible

    const float* brow = sp + HW_ + (size_t)h * W_;                 // b plane row
    const float* rrow = rnd + (size_t)bi * HW_ + (size_t)h * W_;
    const float* drow = dropr + (size_t)h * W_;
    float* osr = out + (size_t)bi * PLANE2_ + (size_t)h * W_;      // state s row
    float* obr = osr + HW_;                                        // state b row

    float e_loc = 0.0f, m_loc = 0.0f;

    #pragma unroll
    for (int c = 0; c < 2; ++c) {
        const int w0 = (c * 256 + tid) * 4;
        const v4f s  = *(const v4f*)&ls[1][w0];
        const v4f up = *(const v4f*)&ls[0][w0];
        const v4f dn = *(const v4f*)&ls[2][w0];
        const float sl = ls[1][(w0 + W_ - 1) & (W_ - 1)];    // left of s.x (wrap)
        const float sr = ls[1][(w0 + 4) & (W_ - 1)];         // right of s.w (wrap)

        // Streamed operands: no reuse -> non-temporal to keep L2 for s/drop.
        const v4f bb = __builtin_nontemporal_load((const v4f*)(brow + w0));
        const v4f rr = __builtin_nontemporal_load((const v4f*)(rrow + w0));
        const v4f dd = *(const v4f*)(drow + w0);             // reused across batches

        const float ns[4] = { up.x + dn.x + sl  + s.y,
                              up.y + dn.y + s.x + s.z,
                              up.z + dn.z + s.y + s.w,
                              up.w + dn.w + s.z + sr };
        const float sv[4] = { s.x, s.y, s.z, s.w };
        const float bv[4] = { bb.x, bb.y, bb.z, bb.w };
        const float rv[4] = { rr.x, rr.y, rr.z, rr.w };
        const float dv[4] = { dd.x, dd.y, dd.z, dd.w };
        float ov[4];

        #pragma unroll
        for (int i = 0; i < 4; ++i) {
            const float sn = sv[i] * ns[i];
            const float d  = 2.0f * sn;          // delta_e
            e_loc -= 0.25f * sn;                 // sum of E/4 = -0.25*s*nsum
            m_loc += sv[i];
            const float p = (d <= 0.0f) ? 1.0f : __expf(-d * bv[i]);
            const bool flip = (rv[i] < p) && (dv[i] > 0.9f);
            ov[i] = flip ? -sv[i] : sv[i];
        }
        v4f o;
        o.x = ov[0]; o.y = ov[1]; o.z = ov[2]; o.w = ov[3];
        __builtin_nontemporal_store(o,  (v4f*)(osr + w0));
        __builtin_nontemporal_store(bb, (v4f*)(obr + w0));   // pass-through b
    }

    // Deterministic fixed-order block reduction of (e, m) partials.
    rede[tid] = e_loc;
    redm[tid] = m_loc;
    __syncthreads();
    for (int s = 128; s > 0; s >>= 1) {
        if (tid < s) {
            rede[tid] += rede[tid + s];
            redm[tid] += redm[tid + s];
        }
        __syncthreads();
    }
    if (tid == 0) {
        partials[(size_t)blk * 2 + 0] = rede[0];
        partials[(size_t)blk * 2 + 1] = redm[0];
    }
}

// ---- observable reduction: one wave per batch, WMMA ones-matmul trick ------
// D = A x ones + C sums 64 f32 partials per v_wmma_f32_16x16x4_f32 step,
// chaining the 16x16 accumulator; 32 steps cover the 2048 row-partials.
__global__ __launch_bounds__(32) void obvs_reduce_kernel(
    const float* __restrict__ partials, float* __restrict__ obvs)
{
    const int bi   = blockIdx.x;
    const int lane = threadIdx.x;

    v2f onesB; onesB[0] = 1.0f; onesB[1] = 1.0f;
    v8f ce, cm;
    #pragma unroll
    for (int j = 0; j < 8; ++j) { ce[j] = 0.0f; cm[j] = 0.0f; }

    const float* base = partials + (size_t)bi * (2 * H_);
    for (int it = 0; it < H_ / 64; ++it) {
        const int i = it * 64 + lane * 2;        // each lane feeds 2 A-slots
        v2f ae, am;
        ae[0] = base[2 * i + 0];
        ae[1] = base[2 * (i + 1) + 0];
        am[0] = base[2 * i + 1];
        am[1] = base[2 * (i + 1) + 1];
        ce = __builtin_amdgcn_wmma_f32_16x16x4_f32(false, ae, false, onesB,
                                                   (short)0, ce, false, false);
        cm = __builtin_amdgcn_wmma_f32_16x16x4_f32(false, am, false, onesB,
                                                   (short)0, cm, false, false);
    }
    // Column L of D holds row-sums for M = lane's half; fold M=0..7 + M=8..15.
    float pe = 0.0f, pm = 0.0f;
    #pragma unroll
    for (int j = 0; j < 8; ++j) { pe += ce[j]; pm += cm[j]; }
    pe += __shfl_xor(pe, 16, 32);
    pm += __shfl_xor(pm, 16, 32);

    if (lane == 0) {
        const float e    = pe;
        const float mean = pm * NINV;
        obvs[0 * B_ + bi] = e;
        obvs[1 * B_ + bi] = e * e;
        obvs[2 * B_ + bi] = fabsf(mean);
        obvs[3 * B_ + bi] = mean * mean;
    }
}

extern "C" void kernel_launch(void* const* d_in, const int* in_sizes, int n_in,
                              void* d_out, int out_size, void* d_ws, size_t ws_size,
                              hipStream_t stream) {
    const float* x     = (const float*)d_in[0];   // (8, 2, 2048, 2048) f32
    const float* rnd   = (const float*)d_in[1];   // (8, 1, 2048, 2048) f32
    const float* dropr = (const float*)d_in[2];   // (2048, 2048) f32
    float* out = (float*)d_out;                   // state (67,108,864) + obvs (32)
    float* partials = (float*)d_ws;               // 16384 * 2 floats = 128 KB

    ising_row_kernel<<<B_ * H_, 256, 0, stream>>>(x, rnd, dropr, out, partials);
    obvs_reduce_kernel<<<B_, 32, 0, stream>>>(partials, out + STATE_ELEMS);
}